// EvolutionModel_91173565759692
// MI455X (gfx1250) — compile-verified
//
#include <hip/hip_runtime.h>
#include <cstdint>

// EvolutionModel: gradient-index ray marching, fully fused, single wave per block.
//   phase 0: async DMA z_vals tile global->LDS            (ASYNCcnt, b128, coalesced)
//   phase 1: 100-step leapfrog + merge-sweep sampling, results staged in LDS
//   phase 2: async DMA LDS->global write-out               (ASYNCcnt, b128, coalesced)

#define N_SAMP   64
#define N_STEPS  100
#define BLOCK    32          // one wave32 per workgroup: no barriers needed
#define ZSTRIDE  68          // floats/ray z staging: 64+4 pad, 272B (16B aligned, <=2-way banks)
#define OSTRIDE  196         // floats/ray out staging: 192+4 pad, 784B (16B aligned, <=2-way banks)

// raw 32-bit LDS offset of a generic pointer to LDS; the ptrtoint also CAPTURES the
// LDS object so "memory"-clobbered asm is known to alias it (keeps DS ops alive).
__device__ __forceinline__ uint32_t lds_off(const void* p) {
    return (uint32_t)(uintptr_t)p;
}

__global__ __launch_bounds__(BLOCK)
void gin_raymarch(const float* __restrict__ x0,
                  const float* __restrict__ v0,
                  const float* __restrict__ zv,
                  const float* __restrict__ ior_c,
                  const float* __restrict__ ior_a,
                  float* __restrict__ out,
                  int n_rays)
{
    __shared__ float lds_z[BLOCK * ZSTRIDE];     //  8704 B
    __shared__ float lds_out[BLOCK * OSTRIDE];   // 25088 B

    const int tid = threadIdx.x;
    const int ray = blockIdx.x * BLOCK + tid;

    // ---- Phase 0: async-stage this block's z_vals (32 rays x 64 f32) into LDS.
    // Lane q moves 16 contiguous bytes; consecutive lanes -> consecutive chunks.
    {
        const uint64_t gz = (uint64_t)(uintptr_t)zv
                          + (uint64_t)blockIdx.x * (uint64_t)(BLOCK * N_SAMP * 4u);
#pragma unroll
        for (int it = 0; it < 16; ++it) {
            const uint32_t q     = (uint32_t)(it * BLOCK + tid);   // 16B-chunk id
            const uint32_t r     = q >> 4;                         // local ray
            const uint32_t c     = q & 15u;                        // chunk within ray
            const uint32_t laddr = lds_off(&lds_z[r * ZSTRIDE]) + c * 16u;
            const uint32_t goff  = q * 16u;
            asm volatile("global_load_async_to_lds_b128 %0, %1, %2"
                         :: "v"(laddr), "v"(goff), "s"(gz) : "memory");
        }
    }
    asm volatile("s_wait_asynccnt 0" ::: "memory");   // single wave: no barrier needed

    // ---- Phase 1: fused leapfrog + merge-sweep sampling (one ray per lane) ----
    if (ray < n_rays) {
        const float cx = ior_c[0], cy = ior_c[1], cz = ior_c[2];
        const float A  = ior_a[0];
        const float dt = 0.02f, half_dt = 0.01f;
        const float inv2s2 = 2.0f;   // 1/(2*sigma^2), sigma = 0.5
        const float invs2  = 4.0f;   // 1/sigma^2

        float px = x0[ray * 3 + 0], py = x0[ray * 3 + 1], pz = x0[ray * 3 + 2];
        float vx = v0[ray * 3 + 0], vy = v0[ray * 3 + 1], vz = v0[ray * 3 + 2];
        const float ox = px, oy = py, oz = pz;       // x_hist[0]
        float dist = 0.0f;
        int s = 0;
        const float* zl = &lds_z[tid * ZSTRIDE];
        float*       ob = &lds_out[tid * OSTRIDE];

        for (int step = 0; step < N_STEPS; ++step) {
            float d0x = px - cx, d0y = py - cy, d0z = pz - cz;
            float g  = __expf(-inv2s2 * (d0x * d0x + d0y * d0y + d0z * d0z));
            float Ag = A * g;
            float k0 = -half_dt * invs2 * Ag * (1.0f + Ag);
            float vhx = vx + k0 * d0x, vhy = vy + k0 * d0y, vhz = vz + k0 * d0z;

            float nx = px + vhx * dt, ny = py + vhy * dt, nz = pz + vhz * dt;
            float sx = nx - px, sy = ny - py, sz = nz - pz;
            float dlen  = __builtin_amdgcn_sqrtf(sx * sx + sy * sy + sz * sz);
            float ndist = dist + dlen;

            float d1x = nx - cx, d1y = ny - cy, d1z = nz - cz;
            float g2  = __expf(-inv2s2 * (d1x * d1x + d1y * d1y + d1z * d1z));
            float Ag2 = A * g2;
            float k1  = -half_dt * invs2 * Ag2 * (1.0f + Ag2);
            vx = vhx + k1 * d1x; vy = vhy + k1 * d1y; vz = vhz + k1 * d1z;

            // consume samples bracketed by (dist, ndist]: c0 = x_step, c1 = x_{step+1}
            while (s < N_SAMP) {
                float zc = zl[s];
                if (zc > ndist) break;
                float mx = nx - px, my = ny - py, mz = nz - pz;
                float inv = __builtin_amdgcn_rsqf(mx * mx + my * my + mz * mz);
                float t = (zc - dist) * inv;
                ob[3 * s + 0] = px + t * mx;
                ob[3 * s + 1] = py + t * my;
                ob[3 * s + 2] = pz + t * mz;
                ++s;
            }
            px = nx; py = ny; pz = nz; dist = ndist;
        }
        // leftovers (z beyond total path): reference -> idx_pos = last, idx_neg = 0
        for (; s < N_SAMP; ++s) {
            float zc = zl[s];
            float mx = ox - px, my = oy - py, mz = oz - pz;
            float inv = __builtin_amdgcn_rsqf(mx * mx + my * my + mz * mz);
            float t = (zc - dist) * inv;
            ob[3 * s + 0] = px + t * mx;
            ob[3 * s + 1] = py + t * my;
            ob[3 * s + 2] = pz + t * mz;
        }
    }

    // ---- Phase 2: async-stream staged results LDS -> global, fully coalesced.
    // Global: 32 rays * 192 f32 contiguous = 1536 16B-chunks, 48 per ray.
    asm volatile("s_wait_dscnt 0" ::: "memory");      // DS writes committed before DMA reads
    {
        const uint64_t go = (uint64_t)(uintptr_t)out
                          + (uint64_t)blockIdx.x * (uint64_t)(BLOCK * N_SAMP * 3 * 4u);
        const uint32_t limit = (uint32_t)((n_rays - blockIdx.x * BLOCK) * 48); // chunks in range
#pragma unroll 4
        for (int it = 0; it < 48; ++it) {
            const uint32_t q = (uint32_t)(it * BLOCK + tid);   // coalesced chunk id
            if (q < limit) {
                const uint32_t r     = q / 48u;                // local ray
                const uint32_t j     = q % 48u;                // chunk within ray
                const uint32_t laddr = lds_off(&lds_out[r * OSTRIDE]) + j * 16u;
                const uint32_t goff  = q * 16u;
                asm volatile("global_store_async_from_lds_b128 %0, %1, %2"
                             :: "v"(goff), "v"(laddr), "s"(go) : "memory");
            }
        }
    }
    // s_endpgm performs an implicit wait-idle, covering outstanding ASYNCcnt stores.
}

extern "C" void kernel_launch(void* const* d_in, const int* in_sizes, int n_in,
                              void* d_out, int out_size, void* d_ws, size_t ws_size,
                              hipStream_t stream)
{
    const float* x0 = (const float*)d_in[0];   // [N,3]
    const float* v0 = (const float*)d_in[1];   // [N,3]
    const float* zv = (const float*)d_in[2];   // [N,64,1] sorted
    const float* c  = (const float*)d_in[3];   // [3]
    const float* A  = (const float*)d_in[4];   // [1]
    float* out = (float*)d_out;                // [N,64,3]

    const int n_rays = in_sizes[0] / 3;
    const int grid   = (n_rays + BLOCK - 1) / BLOCK;
    hipLaunchKernelGGL(gin_raymarch, dim3(grid), dim3(BLOCK), 0, stream,
                       x0, v0, zv, c, A, out, n_rays);
}